// VectorQuantiser_44684839748363
// MI455X (gfx1250) — compile-verified
//
#include <hip/hip_runtime.h>
#include <hip/hip_bf16.h>
#include <math.h>

typedef __attribute__((ext_vector_type(2))) float v2f;
typedef __attribute__((ext_vector_type(8))) float v8f;

#define N_TOK   65536      // 16 * 64 * 64 tokens
#define K_CODE  1024
#define DIM     64
#define CH_STRIDE 4096     // h*w = 64*64 (feature stride inside one image)
#define IMG_STRIDE 262144  // c*h*w per image

// output layout (flat concat in reference return order)
#define OFF_ZQ    0
#define OFF_LOSS  4194304
#define OFF_PERP  4194305
#define OFF_NEWE  4194306
#define OFF_EPROB 4259842
#define OFF_ENC   4260866

__device__ __forceinline__ unsigned int f32_order(float f) {
    unsigned int u = __float_as_uint(f);
    return (u & 0x80000000u) ? ~u : (u | 0x80000000u);
}

// ---------------- init scratch (must re-run every launch) ----------------
__global__ void vq_init(unsigned long long* colbest, int* counts, float* lossAcc) {
    int i = blockIdx.x * blockDim.x + threadIdx.x;
    if (i < K_CODE) { colbest[i] = 0ull; counts[i] = 0; }
    if (i == 0) *lossAcc = 0.0f;
}

// ---------------- kernel 1: WMMA distance GEMM + dual argmax -------------
__global__ __launch_bounds__(128) void vq_main(
        const float* __restrict__ z, const float* __restrict__ E,
        int* __restrict__ idx_out, float* __restrict__ enc_out,
        int* __restrict__ counts, unsigned long long* __restrict__ colbest)
{
    __shared__ unsigned long long shBest[K_CODE];
    for (int i = threadIdx.x; i < K_CODE; i += blockDim.x) shBest[i] = 0ull;
    __syncthreads();

    const int lane = threadIdx.x & 31;
    const int wave = threadIdx.x >> 5;
    const int tokBase = blockIdx.x * 64 + wave * 16;
    const int m    = lane & 15;       // token row within tile / code col within tile
    const int half = lane >> 4;       // 0: K pair {0,1}, 1: K pair {2,3}
    const int o    = half * 2;

    // ---- load A operand: token (tokBase+m), features 4j+o, 4j+o+1 ----
    const int tok  = tokBase + m;
    const int bImg = tok >> 12;
    const int pos  = tok & 4095;
    const float* zb = z + (size_t)bImg * IMG_STRIDE + pos;

    float a[32];
    float zsq = 0.0f;
    #pragma unroll
    for (int j = 0; j < 16; ++j) {
        float x0 = zb[(size_t)(4 * j + o) * CH_STRIDE];
        float x1 = zb[(size_t)(4 * j + o + 1) * CH_STRIDE];
        a[2 * j] = x0; a[2 * j + 1] = x1;
        zsq += x0 * x0 + x1 * x1;
    }
    zsq += __shfl_xor(zsq, 16, 32);   // full ||z_m||^2 (halves are complementary)

    // broadcast ||z||^2 of the 8 rows this lane's C-VGPRs correspond to
    float zn2[8];
    #pragma unroll
    for (int r = 0; r < 8; ++r)
        zn2[r] = __shfl(zsq, r + 8 * half, 32);

    float runmax[8]; int runidx[8];
    #pragma unroll
    for (int r = 0; r < 8; ++r) { runmax[r] = -3.4e38f; runidx[r] = 0; }

    for (int kt = 0; kt < 64; ++kt) {
        const int code = kt * 16 + m;
        const float* erow = E + code * DIM;
        if (kt + 1 < 64)
            __builtin_prefetch(E + (kt + 1) * 16 * DIM + m * DIM, 0, 1);

        // ---- load B operand (codes) + ||e||^2 ----
        float br[32]; float esq = 0.0f;
        #pragma unroll
        for (int j = 0; j < 16; ++j) {
            v2f t = *(const v2f*)(erow + 4 * j + o);
            br[2 * j] = t.x; br[2 * j + 1] = t.y;
            esq += t.x * t.x + t.y * t.y;
        }
        esq += __shfl_xor(esq, 16, 32);

        // ---- two independent 8-deep WMMA chains over d=64 (2x ILP) ----
        v8f acc0 = {0.f, 0.f, 0.f, 0.f, 0.f, 0.f, 0.f, 0.f};
        v8f acc1 = {0.f, 0.f, 0.f, 0.f, 0.f, 0.f, 0.f, 0.f};
        #pragma unroll
        for (int j = 0; j < 8; ++j) {
            v2f av0; av0.x = a[2 * j];       av0.y = a[2 * j + 1];
            v2f bv0; bv0.x = br[2 * j];      bv0.y = br[2 * j + 1];
            acc0 = __builtin_amdgcn_wmma_f32_16x16x4_f32(
                       false, av0, false, bv0, (short)0, acc0, false, false);
            v2f av1; av1.x = a[16 + 2 * j];  av1.y = a[16 + 2 * j + 1];
            v2f bv1; bv1.x = br[16 + 2 * j]; bv1.y = br[16 + 2 * j + 1];
            acc1 = __builtin_amdgcn_wmma_f32_16x16x4_f32(
                       false, av1, false, bv1, (short)0, acc1, false, false);
        }

        // ---- row argmax (over codes) + column argmax (over tokens) ----
        float bestCol = -3.4e38f; int bestRow = 0;
        #pragma unroll
        for (int r = 0; r < 8; ++r) {
            float dot2 = 2.0f * (acc0[r] + acc1[r]);
            float rv = dot2 - esq;                 // -||e||^2 + 2 z.e
            if (rv > runmax[r]) { runmax[r] = rv; runidx[r] = code; }
            float cv = dot2 - zn2[r];              // -||z||^2 + 2 z.e
            if (cv > bestCol) { bestCol = cv; bestRow = r; }
        }
        unsigned int gtok = (unsigned int)(tokBase + bestRow + 8 * half);
        unsigned long long key =
            ((unsigned long long)f32_order(bestCol) << 32) | (0xFFFFFFFFu - gtok);
        unsigned long long other = __shfl_xor(key, 16, 32);
        if (other > key) key = other;
        if (half == 0) atomicMax(&shBest[code], key);   // ds_max_u64
    }

    // ---- finalize row argmax: butterfly within each 16-lane half ----
    #pragma unroll
    for (int r = 0; r < 8; ++r) {
        float v = runmax[r]; int id = runidx[r];
        #pragma unroll
        for (int off = 1; off < 16; off <<= 1) {
            float v2 = __shfl_xor(v, off, 32);
            int   i2 = __shfl_xor(id, off, 32);
            if (v2 > v || (v2 == v && i2 < id)) { v = v2; id = i2; }
        }
        runmax[r] = v; runidx[r] = id;
    }
    if (m == 0) {
        #pragma unroll
        for (int r = 0; r < 8; ++r) {
            int t = tokBase + r + 8 * half;
            idx_out[t] = runidx[r];
            enc_out[t] = (float)runidx[r];
            atomicAdd(&counts[runidx[r]], 1);
        }
    }

    __syncthreads();
    for (int c = threadIdx.x; c < K_CODE; c += blockDim.x) {
        unsigned long long k = shBest[c];
        if (k) atomicMax(&colbest[c], k);
    }
}

// ---------------- kernel 2: z_q gather (b,c,h,w) + loss ----------------
__global__ __launch_bounds__(256) void vq_zq_loss(
        const float* __restrict__ z, const float* __restrict__ E,
        const int* __restrict__ idx, float* __restrict__ zq_out,
        float* __restrict__ lossAcc)
{
    __shared__ float sred[256];
    int e = blockIdx.x * 256 + threadIdx.x;   // e = b*262144 + c*4096 + pos
    int b   = e >> 18;
    int c   = (e >> 12) & 63;
    int pos = e & 4095;
    int tok = (b << 12) | pos;
    float q  = E[idx[tok] * DIM + c];
    float zv = z[e];
    zq_out[e] = q;                            // straight-through == z_q numerically
    float d = q - zv;
    sred[threadIdx.x] = d * d;
    __syncthreads();
    for (int s = 128; s > 0; s >>= 1) {
        if (threadIdx.x < s) sred[threadIdx.x] += sred[threadIdx.x + s];
        __syncthreads();
    }
    if (threadIdx.x == 0) atomicAdd(lossAcc, sred[0]);
}

// ---------------- kernel 3: perplexity, EMA, closest-anchor update ------
__global__ __launch_bounds__(1024) void vq_finalize(
        const float* __restrict__ z, const float* __restrict__ E,
        const float* __restrict__ embed_prob, const int* __restrict__ counts,
        const unsigned long long* __restrict__ colbest,
        const float* __restrict__ lossAcc, float* __restrict__ out)
{
    __shared__ float sred[1024];
    int k = threadIdx.x;
    float p = (float)counts[k] * (1.0f / (float)N_TOK);
    sred[k] = p * logf(p + 1e-10f);
    __syncthreads();
    for (int s = 512; s > 0; s >>= 1) {
        if (k < s) sred[k] += sred[k + s];
        __syncthreads();
    }
    if (k == 0) {
        out[OFF_PERP] = expf(-sred[0]);
        out[OFF_LOSS] = lossAcc[0] * (1.0f + 0.25f) * (1.0f / 4194304.0f);
    }
    float epn = embed_prob[k] * 0.99f + 0.01f * p;
    out[OFF_EPROB + k] = epn;

    unsigned long long key = colbest[k];
    unsigned int tok = 0xFFFFFFFFu - (unsigned int)(key & 0xFFFFFFFFull);
    if (key == 0ull) tok = 0;
    int b = tok >> 12, pos = tok & 4095;

    // exp(-(epn * NUM_EMBED * 10) / (1 - DECAY) - 1e-3)
    float decay = expf(-(epn * 1024.0f * 10.0f) * (1.0f / 0.01f) - 1e-3f);
    const float* zb = z + (size_t)b * IMG_STRIDE + pos;
    for (int c = 0; c < DIM; ++c) {
        float ev = E[k * DIM + c];
        float rf = zb[(size_t)c * CH_STRIDE];
        out[OFF_NEWE + k * DIM + c] = ev * (1.0f - decay) + rf * decay;
    }
}

extern "C" void kernel_launch(void* const* d_in, const int* in_sizes, int n_in,
                              void* d_out, int out_size, void* d_ws, size_t ws_size,
                              hipStream_t stream) {
    const float* z  = (const float*)d_in[0];   // (16,64,64,64)
    const float* E  = (const float*)d_in[1];   // (1024,64)
    const float* ep = (const float*)d_in[2];   // (1024,)
    float* out = (float*)d_out;

    char* ws = (char*)d_ws;
    unsigned long long* colbest = (unsigned long long*)ws;        // 8 KB
    int*   counts  = (int*)(ws + 8192);                           // 4 KB
    int*   idx     = (int*)(ws + 8192 + 4096);                    // 256 KB
    float* lossAcc = (float*)(ws + 8192 + 4096 + 262144);         // 4 B

    vq_init<<<(K_CODE + 255) / 256, 256, 0, stream>>>(colbest, counts, lossAcc);
    vq_main<<<N_TOK / 64, 128, 0, stream>>>(z, E, idx, out + OFF_ENC, counts, colbest);
    vq_zq_loss<<<4194304 / 256, 256, 0, stream>>>(z, E, idx, out + OFF_ZQ, lossAcc);
    vq_finalize<<<1, 1024, 0, stream>>>(z, E, ep, counts, colbest, lossAcc, out);
}